// JobEmbedding_8022998908984
// MI455X (gfx1250) — compile-verified
//
#include <hip/hip_runtime.h>

typedef __attribute__((ext_vector_type(2))) float v2f;
typedef __attribute__((ext_vector_type(8))) float v8f;

constexpr int HID       = 128;
constexpr int N_JOB     = 100000;
constexpr int N_STATION = 20000;
constexpr int N_MACHINE = 20000;
constexpr int N_ROBOT   = 10000;
constexpr int NEDGE     = 800000;

// ---------------------------------------------------------------------------
// prep: Wr_sum = sum of 5 Wr, bl_sum = sum of 5 bl, zero the edge counters
// ---------------------------------------------------------------------------
__global__ void prep_kernel(const float* __restrict__ w0, const float* __restrict__ w1,
                            const float* __restrict__ w2, const float* __restrict__ w3,
                            const float* __restrict__ w4,
                            const float* __restrict__ b0, const float* __restrict__ b1,
                            const float* __restrict__ b2, const float* __restrict__ b3,
                            const float* __restrict__ b4,
                            float* __restrict__ wsum, float* __restrict__ bsum,
                            int* __restrict__ cnt, int ncnt) {
  int i = blockIdx.x * blockDim.x + threadIdx.x;
  if (i < ncnt) cnt[i] = 0;
  if (i < HID * HID) wsum[i] = w0[i] + w1[i] + w2[i] + w3[i] + w4[i];
  if (i < HID) bsum[i] = b0[i] + b1[i] + b2[i] + b3[i] + b4[i];
}

__global__ void count_kernel(const int* __restrict__ dst, int* __restrict__ cnt, int n) {
  int i = blockIdx.x * blockDim.x + threadIdx.x;
  if (i < n) atomicAdd(&cnt[dst[i]], 1);
}

__global__ void inv_kernel(const int* __restrict__ cnt, float* __restrict__ inv, int n) {
  int i = blockIdx.x * blockDim.x + threadIdx.x;
  if (i < n) inv[i] = 1.0f / fmaxf((float)cnt[i], 1.0f);
}

// ---------------------------------------------------------------------------
// C[n][j] = sum_k X[n][k] * W[j][k]  (+ bias[j])     (C = X @ W^T + b)
// X:[N][128]  W:[128][128] row-major  C:[N][128]     N multiple of 16
// Block = 256 threads = 8 waves; wave w owns a 16-row strip x 128 cols.
// B fragments are pre-swizzled into LDS in *exact WMMA fragment order*:
//   ldsF[(kb*8 + t)*32 + lane] = { W[16t+(lane&15)][4kb+2*(lane>>4)],
//                                  W[16t+(lane&15)][4kb+2*(lane>>4)+1] }
// so the inner loop is one conflict-free ds_load_b64 per fragment, no selects.
// Uses V_WMMA_F32_16X16X4_F32: D(16x16,f32) = A(16x4,f32) x B(4x16,f32) + C.
// ---------------------------------------------------------------------------
__global__ __launch_bounds__(256) void gemm_xwt_kernel(
    const float* __restrict__ X, const float* __restrict__ W,
    const float* __restrict__ bias, float* __restrict__ C, int N) {
  __shared__ v2f ldsF[32 * 8 * 32];  // 64 KB, fragment-order B operands

  // Fill: idx -> (j = output col, r = 2*kb + h). Each thread sweeps one W row
  // sequentially (8 B steps, L0-friendly); LDS write has only 2-way conflicts.
  for (int idx = threadIdx.x; idx < 32 * 8 * 32; idx += 256) {
    int j  = idx & 127;        // 0..127 output column
    int r  = idx >> 7;         // 0..63  = 2*kb + h
    int kb = r >> 1;
    int h  = r & 1;
    int t  = j >> 4;
    int lm = j & 15;
    ldsF[(kb * 8 + t) * 32 + h * 16 + lm] = *(const v2f*)(W + j * HID + 2 * r);
  }
  __syncthreads();

  const int wave = threadIdx.x >> 5;
  const int lane = threadIdx.x & 31;
  const int half = lane >> 4;   // 0: lanes 0-15, 1: lanes 16-31
  const int lm   = lane & 15;
  const int rowBase = (blockIdx.x * 8 + wave) * 16;
  if (rowBase >= N) return;     // wave-uniform skip (after barrier)

  v8f acc[8];
#pragma unroll
  for (int t = 0; t < 8; ++t) {
    float b = bias ? bias[t * 16 + lm] : 0.0f;
#pragma unroll
    for (int r = 0; r < 8; ++r) acc[t][r] = b;
  }

  // A fragment (16x4 f32): lane holds row M=lane%16; VGPR0=K(2*half), VGPR1=K(2*half+1)
  const float* xrow = X + (size_t)(rowBase + lm) * HID;

#pragma unroll 4
  for (int kb = 0; kb < 32; ++kb) {
    v2f a = *(const v2f*)(xrow + kb * 4 + half * 2);
    const v2f* bp = &ldsF[kb * 256 + lane];
#pragma unroll
    for (int t = 0; t < 8; ++t) {
      v2f b = bp[t * 32];   // ds_load_b64, 256 contiguous bytes across wave
      acc[t] = __builtin_amdgcn_wmma_f32_16x16x4_f32(
          /*neg_a=*/false, a, /*neg_b=*/false, b,
          /*c_mod=*/(short)0, acc[t], /*reuse_a=*/false, /*reuse_b=*/false);
    }
  }

  // D layout: VGPR r <-> row M = r + 8*half, col N = lane%16
  const int mBase = rowBase + half * 8;
#pragma unroll
  for (int t = 0; t < 8; ++t) {
#pragma unroll
    for (int r = 0; r < 8; ++r) {
      C[(size_t)(mBase + r) * HID + t * 16 + lm] = acc[t][r];
    }
  }
}

// ---------------------------------------------------------------------------
// scatter: out[dst[e]] += y[src[e]] * inv[dst[e]]   (32 threads x float4 per edge)
// unsafeAtomicAdd guarantees hardware global_atomic_add_f32 (no CAS loop).
// ---------------------------------------------------------------------------
__global__ __launch_bounds__(256) void scatter_kernel(
    const float* __restrict__ y, const int* __restrict__ src,
    const int* __restrict__ dst, const float* __restrict__ inv,
    float* __restrict__ out, int nE) {
  int i = blockIdx.x * blockDim.x + threadIdx.x;
  int e = i >> 5;
  if (e >= nE) return;
  int c = (i & 31) * 4;
  int s = src[e];
  int d = dst[e];
  float w = inv[d];
  float4 v = *(const float4*)(y + (size_t)s * HID + c);
  float* op = out + (size_t)d * HID + c;
  unsafeAtomicAdd(op + 0, v.x * w);
  unsafeAtomicAdd(op + 1, v.y * w);
  unsafeAtomicAdd(op + 2, v.z * w);
  unsafeAtomicAdd(op + 3, v.w * w);
}

__global__ void relu_kernel(float* __restrict__ out, int n) {
  int i = blockIdx.x * blockDim.x + threadIdx.x;
  if (i < n) out[i] = fmaxf(out[i], 0.0f);
}

// ---------------------------------------------------------------------------
extern "C" void kernel_launch(void* const* d_in, const int* in_sizes, int n_in,
                              void* d_out, int out_size, void* d_ws, size_t ws_size,
                              hipStream_t stream) {
  (void)in_sizes; (void)n_in; (void)out_size; (void)ws_size;

  const float* x_job     = (const float*)d_in[0];
  const float* x_station = (const float*)d_in[1];
  const float* x_machine = (const float*)d_in[2];
  const float* x_robot   = (const float*)d_in[3];

  const int*   src[5];
  const int*   dst[5];
  const float* Wl[5];
  const float* bl[5];
  const float* Wr[5];
  for (int r = 0; r < 5; ++r) {
    src[r] = (const int*)  d_in[4 + 5 * r + 0];
    dst[r] = (const int*)  d_in[4 + 5 * r + 1];
    Wl[r]  = (const float*)d_in[4 + 5 * r + 2];
    bl[r]  = (const float*)d_in[4 + 5 * r + 3];
    Wr[r]  = (const float*)d_in[4 + 5 * r + 4];
  }
  const float* xsrc[5] = {x_station, x_station, x_machine, x_machine, x_robot};
  const int    nsrc[5] = {N_STATION, N_STATION, N_MACHINE, N_MACHINE, N_ROBOT};

  // workspace layout
  float* wsum = (float*)d_ws;             // 128*128
  float* bsum = wsum + HID * HID;         // 128
  int*   cnt  = (int*)(bsum + HID);       // 5*N_JOB
  float* inv  = (float*)(cnt + 5 * N_JOB);// 5*N_JOB
  float* y[5];
  {
    float* p = inv + 5 * N_JOB;
    for (int r = 0; r < 5; ++r) { y[r] = p; p += (size_t)nsrc[r] * HID; }
  }
  float* out = (float*)d_out;

  // 1) Wr_sum / bl_sum / zero counters
  prep_kernel<<<(5 * N_JOB + 255) / 256, 256, 0, stream>>>(
      Wr[0], Wr[1], Wr[2], Wr[3], Wr[4],
      bl[0], bl[1], bl[2], bl[3], bl[4],
      wsum, bsum, cnt, 5 * N_JOB);

  // 2) per-relation in-degree counts
  for (int r = 0; r < 5; ++r)
    count_kernel<<<(NEDGE + 255) / 256, 256, 0, stream>>>(dst[r], cnt + r * N_JOB, NEDGE);

  // 3) 1/max(cnt,1)
  inv_kernel<<<(5 * N_JOB + 255) / 256, 256, 0, stream>>>(cnt, inv, 5 * N_JOB);

  // 4) y_r = x_src @ Wl_r^T  (WMMA, per-source-node — 90K rows total instead of 500K)
  for (int r = 0; r < 5; ++r) {
    int blocks = (nsrc[r] / 16 + 7) / 8;
    gemm_xwt_kernel<<<blocks, 256, 0, stream>>>(xsrc[r], Wl[r], nullptr, y[r], nsrc[r]);
  }

  // 5) out = x_job @ (sum Wr)^T + sum bl   (single root GEMM, acc lives in d_out)
  gemm_xwt_kernel<<<(N_JOB / 16 + 7) / 8, 256, 0, stream>>>(x_job, wsum, bsum, out, N_JOB);

  // 6) out[dst] += y_r[src] / cnt_r[dst]   (L2-resident float atomics)
  for (int r = 0; r < 5; ++r)
    scatter_kernel<<<(NEDGE * 32 + 255) / 256, 256, 0, stream>>>(
        y[r], src[r], dst[r], inv + r * N_JOB, out, NEDGE);

  // 7) ReLU in place
  relu_kernel<<<(N_JOB * HID + 255) / 256, 256, 0, stream>>>(out, N_JOB * HID);
}